// DPhysicsEngine_33054068310307
// MI455X (gfx1250) — compile-verified
//
#include <hip/hip_runtime.h>
#include <math.h>

#define NB   1024
#define ND   8
#define NP   128
#define NPTS 1024
#define GW   256  // grid H=W

typedef float v2f __attribute__((ext_vector_type(2)));
typedef float v8f __attribute__((ext_vector_type(8)));

__device__ __forceinline__ void m3_mul(const float* A, const float* B, float* C) {
#pragma unroll
  for (int i = 0; i < 3; ++i)
#pragma unroll
    for (int k = 0; k < 3; ++k)
      C[i*3+k] = A[i*3+0]*B[0*3+k] + A[i*3+1]*B[1*3+k] + A[i*3+2]*B[2*3+k];
}
// C = A * B^T
__device__ __forceinline__ void m3_mul_nt(const float* A, const float* B, float* C) {
#pragma unroll
  for (int i = 0; i < 3; ++i)
#pragma unroll
    for (int l = 0; l < 3; ++l)
      C[i*3+l] = A[i*3+0]*B[l*3+0] + A[i*3+1]*B[l*3+1] + A[i*3+2]*B[l*3+2];
}
__device__ __forceinline__ void m3_vec(const float* A, const float* v, float* o) {
#pragma unroll
  for (int i = 0; i < 3; ++i)
    o[i] = A[i*3+0]*v[0] + A[i*3+1]*v[1] + A[i*3+2]*v[2];
}

__global__ __launch_bounds__(256)
void dphys_step(const float* __restrict__ x, const float* __restrict__ xd,
                const float* __restrict__ q, const float* __restrict__ omega,
                const float* __restrict__ thetas, const float* __restrict__ controls,
                const float* __restrict__ z_grid, const float* __restrict__ z_grid_grad,
                const float* __restrict__ jp,  const float* __restrict__ jlp,
                const float* __restrict__ jlc, const float* __restrict__ dpi,
                const float* __restrict__ dpm, const float* __restrict__ body_cog,
                const float* __restrict__ body_inertia, const float* __restrict__ ddir,
                float* __restrict__ out)
{
  const int b    = blockIdx.x;
  const int tid  = threadIdx.x;
  const int lane = tid & 31;
  const int wv   = tid >> 5;

  // ---- output offsets (reference return-tuple order, all f32) ----
  float* o_nx  = out;
  float* o_nxd = out + (size_t)NB*3;
  float* o_nq  = out + (size_t)NB*6;
  float* o_nom = out + (size_t)NB*10;
  float* o_nth = out + (size_t)NB*13;
  float* o_xdd = out + (size_t)NB*21;
  float* o_omd = out + (size_t)NB*24;
  float* o_thd = out + (size_t)NB*27;
  float* o_fs  = out + (size_t)NB*35;
  float* o_ff  = o_fs + (size_t)NB*NPTS*3;
  float* o_ic  = o_ff + (size_t)NB*NPTS*3;
  float* o_tq  = o_ic + (size_t)NB*NPTS;
  float* o_rp  = o_tq + (size_t)NB*3;
  float* o_tr  = o_rp + (size_t)NB*NPTS*3;

  __shared__ float s_M[ND*9];     // fused M[d] = R_world * rots[d]
  __shared__ float s_t[ND*3];     // t[d] = R_world*jp[d] + x
  __shared__ float s_thr[ND*3];   // world thrust per joint
  __shared__ float s_gdd[3];
  __shared__ float s_gcog[3];
  __shared__ float s_misc[2];     // num_contacts, k_damping
  __shared__ float s_pts[NPTS*3];
  __shared__ float s_n[NPTS*3];
  __shared__ float s_dh[NPTS];
  __shared__ float s_ic[NPTS];
  __shared__ float s_red[256*6];

  // thread-0 persistent state (prologue -> epilogue)
  float Rw[9] = {};
  float inertiaW[9] = {};
  float total_mass = 0.0f;

  // ================= prologue: per-batch small algebra (thread 0) =================
  if (tid == 0) {
    const float xb0 = x[b*3+0], xb1 = x[b*3+1], xb2 = x[b*3+2];
    const float qw = q[b*4+0], qx = q[b*4+1], qy = q[b*4+2], qz = q[b*4+3];
    Rw[0] = 1.f-2.f*(qy*qy+qz*qz); Rw[1] = 2.f*(qx*qy-qw*qz); Rw[2] = 2.f*(qx*qz+qw*qy);
    Rw[3] = 2.f*(qx*qy+qw*qz); Rw[4] = 1.f-2.f*(qx*qx+qz*qz); Rw[5] = 2.f*(qy*qz-qw*qx);
    Rw[6] = 2.f*(qx*qz-qw*qy); Rw[7] = 2.f*(qy*qz+qw*qx); Rw[8] = 1.f-2.f*(qx*qx+qy*qy);

    const float dd0 = ddir[0], dd1 = ddir[1], dd2 = ddir[2];

    total_mass = 40.0f;
#pragma unroll
    for (int d = 0; d < ND; ++d) total_mass += dpm[d];

    float Iov[9] = {0,0,0,0,0,0,0,0,0};
    float cg[3]  = {0,0,0};
#pragma unroll
    for (int d = 0; d < ND; ++d) {
      const float th = thetas[b*ND+d], c = cosf(th), s = sinf(th);
      const float rt[9] = {c,0.f,s, 0.f,1.f,0.f, -s,0.f,c};
      float cl[3];
      m3_vec(rt, jlc + d*3, cl);
      cl[0] += jp[d*3+0]; cl[1] += jp[d*3+1]; cl[2] += jp[d*3+2];
      cg[0] += dpm[d]*cl[0]; cg[1] += dpm[d]*cl[1]; cg[2] += dpm[d]*cl[2];
      float T[9], RI[9];
      m3_mul(rt, dpi + d*9, T);
      m3_mul_nt(T, rt, RI);
#pragma unroll
      for (int i = 0; i < 9; ++i) Iov[i] += RI[i];
    }
    const float inv_tm = 1.0f/total_mass;
    cg[0] = (cg[0] + 40.0f*body_cog[0]) * inv_tm;
    cg[1] = (cg[1] + 40.0f*body_cog[1]) * inv_tm;
    cg[2] = (cg[2] + 40.0f*body_cog[2]) * inv_tm;

#pragma unroll
    for (int d = 0; d < ND; ++d) {
      const float th = thetas[b*ND+d], c = cosf(th), s = sinf(th);
      const float rt[9] = {c,0.f,s, 0.f,1.f,0.f, -s,0.f,c};
      float cl[3];
      m3_vec(rt, jlc + d*3, cl);
      cl[0] += jp[d*3+0]; cl[1] += jp[d*3+1]; cl[2] += jp[d*3+2];
      const float dv[3] = {cl[0]-cg[0], cl[1]-cg[1], cl[2]-cg[2]};
      const float d2 = dv[0]*dv[0]+dv[1]*dv[1]+dv[2]*dv[2];
#pragma unroll
      for (int i = 0; i < 3; ++i)
#pragma unroll
        for (int l = 0; l < 3; ++l)
          Iov[i*3+l] += dpm[d]*(((i==l)?d2:0.f) - dv[i]*dv[l]);
      // fused transform M = Rw*rt, t = Rw*jp + x
      float M[9];
      m3_mul(Rw, rt, M);
#pragma unroll
      for (int i = 0; i < 9; ++i) s_M[d*9+i] = M[i];
      float tv[3];
      m3_vec(Rw, jp + d*3, tv);
      s_t[d*3+0] = tv[0] + xb0; s_t[d*3+1] = tv[1] + xb1; s_t[d*3+2] = tv[2] + xb2;
      // thrust
      const float vc = controls[b*2*ND + d];
      float tl[3] = {(rt[0]*dd0+rt[1]*dd1+rt[2]*dd2)*vc,
                     (rt[3]*dd0+rt[4]*dd1+rt[5]*dd2)*vc,
                     (rt[6]*dd0+rt[7]*dd1+rt[8]*dd2)*vc};
      float tw[3];
      m3_vec(Rw, tl, tw);
      s_thr[d*3+0] = tw[0]; s_thr[d*3+1] = tw[1]; s_thr[d*3+2] = tw[2];
    }
    const float db[3] = {body_cog[0]-cg[0], body_cog[1]-cg[1], body_cog[2]-cg[2]};
    const float db2 = db[0]*db[0]+db[1]*db[1]+db[2]*db[2];
#pragma unroll
    for (int i = 0; i < 3; ++i)
#pragma unroll
      for (int l = 0; l < 3; ++l)
        Iov[i*3+l] += body_inertia[i*3+l] + 40.0f*(((i==l)?db2:0.f) - db[i]*db[l]);
    float T[9];
    m3_mul(Rw, Iov, T);
    m3_mul_nt(T, Rw, inertiaW);

    float gdd[3]; m3_vec(Rw, ddir, gdd);
    s_gdd[0]=gdd[0]; s_gdd[1]=gdd[1]; s_gdd[2]=gdd[2];
    float gc[3]; m3_vec(Rw, cg, gc);
    s_gcog[0]=gc[0]+xb0; s_gcog[1]=gc[1]+xb1; s_gcog[2]=gc[2]+xb2;
  }
  __syncthreads();

  // ================= pass A: WMMA point transform =================
  // D(16x16) = A(16x4) x B(4x16): A rows = [px,py,pz,1], B cols j<3 = [M[j][0..2], t[j]]
  {
    const int col = lane & 15;
    const int hi  = lane >> 4;
    for (int g = wv; g < NPTS/16; g += 8) {
      const int base = g * 16;
      const int d    = base >> 7;         // 128 points per joint
      const float* lp = jlp + (size_t)(base + col) * 3;
      v2f a;
      a.x = hi ? lp[2] : lp[0];
      a.y = hi ? 1.0f  : lp[1];
      float b0 = 0.0f, b1 = 0.0f;
      if (col < 3) {
        b0 = hi ? s_M[d*9 + col*3 + 2] : s_M[d*9 + col*3 + 0];
        b1 = hi ? s_t[d*3 + col]       : s_M[d*9 + col*3 + 1];
      }
      v2f bm; bm.x = b0; bm.y = b1;
      v8f acc = {0.f,0.f,0.f,0.f,0.f,0.f,0.f,0.f};
      acc = __builtin_amdgcn_wmma_f32_16x16x4_f32(
          false, a, false, bm, (short)0, acc, false, false);
      if (col < 3) {
#pragma unroll
        for (int r = 0; r < 8; ++r) {
          const int pi = base + r + 8*hi;     // C/D layout: vgpr r, lanes16-31 -> M=r+8
          const float val = acc[r];
          s_pts[pi*3 + col] = val;
          o_rp[(size_t)b*NPTS*3 + (size_t)pi*3 + col] = val;
        }
      }
    }
  }
  __syncthreads();

  // ================= pass B: grid contact probe =================
  {
    const float* zg  = z_grid + (size_t)b*GW*GW;
    const float* gxg = z_grid_grad + (size_t)b*2*GW*GW;
    const float* gyg = gxg + (size_t)GW*GW;
    float ic_acc = 0.0f;
#pragma unroll
    for (int k = 0; k < 4; ++k) {
      const int p = k*256 + tid;
      const float px = s_pts[p*3+0], py = s_pts[p*3+1], pz = s_pts[p*3+2];
      float u = (px + 6.4f) * (255.0f/12.8f);
      float v = (py + 6.4f) * (255.0f/12.8f);
      u = fminf(fmaxf(u, 0.0f), 255.0f - 1e-5f);
      v = fminf(fmaxf(v, 0.0f), 255.0f - 1e-5f);
      const int u0 = (int)floorf(u), v0 = (int)floorf(v);
      const float fu = u - (float)u0, fv = v - (float)v0;
      const int i00 = v0*GW + u0;
      const float w00 = (1.f-fu)*(1.f-fv), w01 = fu*(1.f-fv), w10 = (1.f-fu)*fv, w11 = fu*fv;
      const float zs = zg[i00]*w00 + zg[i00+1]*w01 + zg[i00+GW]*w10 + zg[i00+GW+1]*w11;
      const float gx = gxg[i00]*w00 + gxg[i00+1]*w01 + gxg[i00+GW]*w10 + gxg[i00+GW+1]*w11;
      const float gy = gyg[i00]*w00 + gyg[i00+1]*w01 + gyg[i00+GW]*w10 + gyg[i00+GW+1]*w11;
      const float nrm = sqrtf(gx*gx + gy*gy + 1.0f);
      const float inv = 1.0f/(nrm + 1e-8f);
      const float nx = -gx*inv, ny = -gy*inv, nz = inv;
      const float dh = (pz - zs) * nz;
      const float ic = 0.5f*(1.0f + tanhf(-dh * (1.7320508075688772f/0.03f)));
      s_n[p*3+0] = nx; s_n[p*3+1] = ny; s_n[p*3+2] = nz;
      s_dh[p] = dh; s_ic[p] = ic;
      o_ic[(size_t)b*NPTS + p] = ic;
      ic_acc += ic;
    }
    s_red[tid] = ic_acc;
  }
  __syncthreads();
  if (tid == 0) {
    float sum = 0.0f;
    for (int t = 0; t < 256; ++t) sum += s_red[t];      // deterministic order
    const float nc = fmaxf(sum, 1.0f);
    s_misc[0] = nc;
    s_misc[1] = sqrtf(total_mass * 5000.0f / nc);       // DAMPING_ALPHA*2 = 1
  }
  __syncthreads();

  // ================= pass C: forces =================
  {
    const float nc = s_misc[0], kd = s_misc[1];
    const float gc0 = s_gcog[0], gc1 = s_gcog[1], gc2 = s_gcog[2];
    const float gd0 = s_gdd[0], gd1 = s_gdd[1], gd2 = s_gdd[2];
    const float xd0 = xd[b*3+0], xd1 = xd[b*3+1], xd2 = xd[b*3+2];
    const float om0 = omega[b*3+0], om1 = omega[b*3+1], om2 = omega[b*3+2];
    float a_acc[3] = {0,0,0}, t_acc[3] = {0,0,0};
#pragma unroll
    for (int k = 0; k < 4; ++k) {
      const int p = k*256 + tid;
      const int d = p >> 7;
      const float nx = s_n[p*3+0], ny = s_n[p*3+1], nz = s_n[p*3+2];
      const float dh = s_dh[p], ic = s_ic[p];
      const float cc0 = s_pts[p*3+0]-gc0, cc1 = s_pts[p*3+1]-gc1, cc2 = s_pts[p*3+2]-gc2;
      const float vp0 = xd0 + om1*cc2 - om2*cc1;
      const float vp1 = xd1 + om2*cc0 - om0*cc2;
      const float vp2 = xd2 + om0*cc1 - om1*cc0;
      const float xdn = vp0*nx + vp1*ny + vp2*nz;
      const float coef = -(5000.0f*dh*ic + kd*xdn) * ic / nc;
      const float fs0 = coef*nx, fs1 = coef*ny, fs2 = coef*nz;
      const float Nmag = sqrtf(fs0*fs0 + fs1*fs1 + fs2*fs2);
      // forward / lateral dirs
      const float gdn = gd0*nx + gd1*ny + gd2*nz;
      float fw0 = gd0-gdn*nx, fw1 = gd1-gdn*ny, fw2 = gd2-gdn*nz;
      {
        const float inv = 1.0f/(sqrtf(fw0*fw0+fw1*fw1+fw2*fw2) + 1e-8f);
        fw0 *= inv; fw1 *= inv; fw2 *= inv;
      }
      float lt0 = fw1*nz - fw2*ny, lt1 = fw2*nx - fw0*nz, lt2 = fw0*ny - fw1*nx;
      {
        const float inv = 1.0f/(sqrtf(lt0*lt0+lt1*lt1+lt2*lt2) + 1e-8f);
        lt0 *= inv; lt1 *= inv; lt2 *= inv;
      }
      const float th0 = s_thr[d*3+0], th1 = s_thr[d*3+1], th2 = s_thr[d*3+2];
      const float dv0 = th0-vp0, dv1 = th1-vp1, dv2 = th2-vp2;
      const float dvn = dv0*nx + dv1*ny + dv2*nz;
      const float t0 = tanhf(dv0 - dvn*nx);
      const float t1 = tanhf(dv1 - dvn*ny);
      const float t2 = tanhf(dv2 - dvn*nz);
      const float lon = t0*fw0 + t1*fw1 + t2*fw2;
      const float lat = t0*lt0 + t1*lt1 + t2*lt2;
      const float s  = 0.5f*Nmag;    // K_LON = K_LAT = 0.5
      const float ff0 = s*(lon*fw0 + lat*lt0);
      const float ff1 = s*(lon*fw1 + lat*lt1);
      const float ff2 = s*(lon*fw2 + lat*lt2);
      const float ac0 = fs0+ff0, ac1 = fs1+ff1, ac2 = fs2+ff2;
      const size_t o3 = (size_t)b*NPTS*3 + (size_t)p*3;
      o_fs[o3+0]=fs0; o_fs[o3+1]=fs1; o_fs[o3+2]=fs2;
      o_ff[o3+0]=ff0; o_ff[o3+1]=ff1; o_ff[o3+2]=ff2;
      o_tr[o3+0]=th0; o_tr[o3+1]=th1; o_tr[o3+2]=th2;
      a_acc[0]+=ac0; a_acc[1]+=ac1; a_acc[2]+=ac2;
      t_acc[0]+=cc1*ac2-cc2*ac1; t_acc[1]+=cc2*ac0-cc0*ac2; t_acc[2]+=cc0*ac1-cc1*ac0;
    }
#pragma unroll
    for (int c = 0; c < 3; ++c) { s_red[tid*6+c]=a_acc[c]; s_red[tid*6+3+c]=t_acc[c]; }
  }
  __syncthreads();

  // ================= epilogue: reductions, solve, integrate (thread 0) =================
  if (tid == 0) {
    float act[3]={0,0,0}, tq[3]={0,0,0};
    for (int t = 0; t < 256; ++t) {
#pragma unroll
      for (int c = 0; c < 3; ++c) { act[c]+=s_red[t*6+c]; tq[c]+=s_red[t*6+3+c]; }
    }
#pragma unroll
    for (int c = 0; c < 3; ++c) tq[c] = fminf(fmaxf(tq[c], -200.0f), 200.0f);
    const float* A = inertiaW;
    const float adj[9] = {
      A[4]*A[8]-A[5]*A[7], A[2]*A[7]-A[1]*A[8], A[1]*A[5]-A[2]*A[4],
      A[5]*A[6]-A[3]*A[8], A[0]*A[8]-A[2]*A[6], A[2]*A[3]-A[0]*A[5],
      A[3]*A[7]-A[4]*A[6], A[1]*A[6]-A[0]*A[7], A[0]*A[4]-A[1]*A[3]};
    const float det = A[0]*adj[0] + A[1]*adj[3] + A[2]*adj[6];
    const float idet = 1.0f/det;
    float omd[3];
#pragma unroll
    for (int i = 0; i < 3; ++i)
      omd[i] = (adj[i*3]*tq[0] + adj[i*3+1]*tq[1] + adj[i*3+2]*tq[2]) * idet;
    const float itm = 1.0f/total_mass;
    const float xdd0 = act[0]*itm, xdd1 = act[1]*itm, xdd2 = (act[2]-total_mass*9.8f)*itm;
    const float xd0 = xd[b*3+0], xd1 = xd[b*3+1], xd2 = xd[b*3+2];
    const float nxd0 = xd0+xdd0*0.01f, nxd1 = xd1+xdd1*0.01f, nxd2 = xd2+xdd2*0.01f;
    const float nx0 = x[b*3+0]+nxd0*0.01f, nx1 = x[b*3+1]+nxd1*0.01f, nx2 = x[b*3+2]+nxd2*0.01f;
    const float no0 = omega[b*3+0]+omd[0]*0.01f;
    const float no1 = omega[b*3+1]+omd[1]*0.01f;
    const float no2 = omega[b*3+2]+omd[2]*0.01f;
    const float qw = q[b*4+0], qx = q[b*4+1], qy = q[b*4+2], qz = q[b*4+3];
    float nq0 = qw + 0.005f*(-qx*no0 - qy*no1 - qz*no2);
    float nq1 = qx + 0.005f*( qw*no0 + qy*no2 - qz*no1);
    float nq2 = qy + 0.005f*( qw*no1 - qx*no2 + qz*no0);
    float nq3 = qz + 0.005f*( qw*no2 + qx*no1 - qy*no0);
    const float qinv = 1.0f/(sqrtf(nq0*nq0+nq1*nq1+nq2*nq2+nq3*nq3) + 1e-8f);
    o_nx[b*3+0]=nx0;  o_nx[b*3+1]=nx1;  o_nx[b*3+2]=nx2;
    o_nxd[b*3+0]=nxd0; o_nxd[b*3+1]=nxd1; o_nxd[b*3+2]=nxd2;
    o_nq[b*4+0]=nq0*qinv; o_nq[b*4+1]=nq1*qinv; o_nq[b*4+2]=nq2*qinv; o_nq[b*4+3]=nq3*qinv;
    o_nom[b*3+0]=no0; o_nom[b*3+1]=no1; o_nom[b*3+2]=no2;
    o_xdd[b*3+0]=xdd0; o_xdd[b*3+1]=xdd1; o_xdd[b*3+2]=xdd2;
    o_omd[b*3+0]=omd[0]; o_omd[b*3+1]=omd[1]; o_omd[b*3+2]=omd[2];
    o_tq[b*3+0]=tq[0]; o_tq[b*3+1]=tq[1]; o_tq[b*3+2]=tq[2];
#pragma unroll
    for (int j = 0; j < ND; ++j) {
      const float td = fminf(fmaxf(controls[b*2*ND + ND + j], -0.5f), 0.5f);
      o_thd[b*ND+j] = td;
      o_nth[b*ND+j] = fminf(fmaxf(thetas[b*ND+j] + td*0.01f, -1.0f), 1.0f);
    }
  }
}

extern "C" void kernel_launch(void* const* d_in, const int* in_sizes, int n_in,
                              void* d_out, int out_size, void* d_ws, size_t ws_size,
                              hipStream_t stream) {
  (void)in_sizes; (void)n_in; (void)out_size; (void)d_ws; (void)ws_size;
  const float* x        = (const float*)d_in[0];
  const float* xd       = (const float*)d_in[1];
  const float* q        = (const float*)d_in[2];
  const float* omega    = (const float*)d_in[3];
  const float* thetas   = (const float*)d_in[4];
  const float* controls = (const float*)d_in[5];
  const float* z_grid   = (const float*)d_in[6];
  const float* z_gg     = (const float*)d_in[7];
  const float* jp       = (const float*)d_in[8];
  const float* jlp      = (const float*)d_in[9];
  const float* jlc      = (const float*)d_in[10];
  const float* dpi      = (const float*)d_in[11];
  const float* dpm      = (const float*)d_in[12];
  const float* bcog     = (const float*)d_in[13];
  const float* binert   = (const float*)d_in[14];
  const float* ddir     = (const float*)d_in[15];
  dphys_step<<<dim3(NB), dim3(256), 0, stream>>>(
      x, xd, q, omega, thetas, controls, z_grid, z_gg,
      jp, jlp, jlc, dpi, dpm, bcog, binert, ddir, (float*)d_out);
}